// GenerativeModel_52544629899928
// MI455X (gfx1250) — compile-verified
//
#include <hip/hip_runtime.h>
#include <stdint.h>

// Problem constants (fixed by the reference)
#define LPATH   2048            // PATH_DIM
#define NSEG    (LPATH - 1)     // 2047 outputs per row
#define CHUNK   64              // segments per lane
#define NCHUNK  32              // lanes per wave (wave32)
#define LDS_STRIDE 65           // 64 dwords + 1 pad dword (TDM pad_enable)
#define LDS_FLOATS (NCHUNK * LDS_STRIDE)  // 2080

typedef unsigned int u32;
typedef __attribute__((ext_vector_type(4))) u32 u32x4;
typedef __attribute__((ext_vector_type(8))) u32 u32x8;

struct Sig { float a1[2]; float a2[4]; float a3[8]; };

__device__ inline Sig sig_zero() {
  Sig s;
#pragma unroll
  for (int i = 0; i < 2; ++i) s.a1[i] = 0.f;
#pragma unroll
  for (int i = 0; i < 4; ++i) s.a2[i] = 0.f;
#pragma unroll
  for (int i = 0; i < 8; ++i) s.a3[i] = 0.f;
  return s;
}

// Chen product: A (earlier path) ⊗ G (later path), truncated at level 3.
__device__ inline Sig chen_comb(const Sig& A, const Sig& G) {
  Sig c;
#pragma unroll
  for (int i = 0; i < 2; ++i) c.a1[i] = A.a1[i] + G.a1[i];
#pragma unroll
  for (int i = 0; i < 2; ++i)
#pragma unroll
    for (int j = 0; j < 2; ++j)
      c.a2[i*2+j] = A.a2[i*2+j] + A.a1[i]*G.a1[j] + G.a2[i*2+j];
#pragma unroll
  for (int i = 0; i < 2; ++i)
#pragma unroll
    for (int j = 0; j < 2; ++j)
#pragma unroll
      for (int k = 0; k < 2; ++k)
        c.a3[(i*2+j)*2+k] = A.a3[(i*2+j)*2+k]
                          + A.a2[i*2+j]*G.a1[k]
                          + A.a1[i]*G.a2[j*2+k]
                          + G.a3[(i*2+j)*2+k];
  return c;
}

// In-place Chen update with one linear segment of increment d=(du,dv):
// segment levels are (d, d⊗d/2, d⊗d⊗d/6).
__device__ inline void chen_seg(Sig& a, float du, float dv) {
  float d[2] = {du, dv};
  float g2[4], g3[8];
#pragma unroll
  for (int i = 0; i < 2; ++i)
#pragma unroll
    for (int j = 0; j < 2; ++j)
      g2[i*2+j] = 0.5f * d[i] * d[j];
#pragma unroll
  for (int i = 0; i < 2; ++i)
#pragma unroll
    for (int j = 0; j < 2; ++j)
#pragma unroll
      for (int k = 0; k < 2; ++k)
        g3[(i*2+j)*2+k] = (1.0f/6.0f) * d[i] * d[j] * d[k];
  // level 3 first (uses old a2, a1)
#pragma unroll
  for (int i = 0; i < 2; ++i)
#pragma unroll
    for (int j = 0; j < 2; ++j)
#pragma unroll
      for (int k = 0; k < 2; ++k)
        a.a3[(i*2+j)*2+k] += a.a2[i*2+j]*d[k] + a.a1[i]*g2[j*2+k] + g3[(i*2+j)*2+k];
#pragma unroll
  for (int i = 0; i < 2; ++i)
#pragma unroll
    for (int j = 0; j < 2; ++j)
      a.a2[i*2+j] += a.a1[i]*d[j] + g2[i*2+j];
#pragma unroll
  for (int i = 0; i < 2; ++i) a.a1[i] += d[i];
}

__device__ inline Sig shfl_up_sig(const Sig& s, int d) {
  Sig r;
#pragma unroll
  for (int i = 0; i < 2; ++i) r.a1[i] = __shfl_up(s.a1[i], d, 32);
#pragma unroll
  for (int i = 0; i < 4; ++i) r.a2[i] = __shfl_up(s.a2[i], d, 32);
#pragma unroll
  for (int i = 0; i < 8; ++i) r.a3[i] = __shfl_up(s.a3[i], d, 32);
  return r;
}

// Pointwise augment MLP: 1 -> 8 (relu) -> 2
__device__ inline float2 mlp2(float xv, const float* w1r, const float* b1r,
                              const float* w2r, const float* b2r) {
  float p0 = b2r[0], p1 = b2r[1];
#pragma unroll
  for (int h = 0; h < 8; ++h) {
    float t = fmaxf(fmaf(w1r[h], xv, b1r[h]), 0.f);
    p0 = fmaf(w2r[h],     t, p0);
    p1 = fmaf(w2r[8 + h], t, p1);
  }
  return make_float2(p0, p1);
}

__global__ __launch_bounds__(32) void sig_scan_kernel(
    const float* __restrict__ x,   const float* __restrict__ w1,
    const float* __restrict__ b1,  const float* __restrict__ w2,
    const float* __restrict__ b2,  const float* __restrict__ wout,
    float* __restrict__ out)
{
  __shared__ float xs[LDS_FLOATS];   // TDM destination (padded: 65 dwords / 64 elems)
  __shared__ float os[LDS_FLOATS];   // output staging (same padded layout)

  const int lane = threadIdx.x;      // 0..31, one chunk per lane
  const int brow = blockIdx.x;       // one batch row per wave/block

  // ---------- TDM: DMA row (8KB) global -> LDS with hardware pad stagger ----------
  {
    u32 lds_off = (u32)(uintptr_t)(&xs[0]);                // flat addr low 32b == LDS byte addr
    uint64_t ga = (uint64_t)(uintptr_t)(x + (size_t)brow * LPATH);
    u32x4 g0;
    g0[0] = 1u;                                            // count=1, user descriptor
    g0[1] = lds_off;                                       // lds_addr (bytes)
    g0[2] = (u32)(ga & 0xffffffffu);                       // global_addr lo
    g0[3] = (u32)((ga >> 32) & 0x01ffffffu) | (2u << 30);  // global_addr hi | type=2
    u32x8 g1;
    g1[0] = (2u << 16) | (1u << 20) | (5u << 22);          // data_size=4B, pad_enable, pad every 64 dw, pad 1 dw
    g1[1] = ((u32)LPATH & 0xffffu) << 16;                  // tensor_dim0 lo16
    g1[2] = ((u32)LPATH >> 16) | (1u << 16);               // tensor_dim0 hi16 | tensor_dim1=1
    g1[3] = ((u32)LPATH) << 16;                            // tensor_dim1 hi16 | tile_dim0=2048
    g1[4] = 1u;                                            // tile_dim1=1, tile_dim2=0
    g1[5] = (u32)LPATH;                                    // tensor_dim0_stride lo32
    g1[6] = 0u;
    g1[7] = 0u;
    asm volatile("tensor_load_to_lds %0, %1" :: "s"(g0), "s"(g1) : "memory");
  }

  // ---------- tiny uniform weights into registers (overlaps the TDM) ----------
  float w1r[8], b1r[8], w2r[16], b2r[2], wo[14];
#pragma unroll
  for (int i = 0; i < 8; ++i)  { w1r[i] = w1[i]; b1r[i] = b1[i]; }
#pragma unroll
  for (int i = 0; i < 16; ++i) w2r[i] = w2[i];
  b2r[0] = b2[0]; b2r[1] = b2[1];
#pragma unroll
  for (int i = 0; i < 14; ++i) wo[i] = wout[i];

  __builtin_amdgcn_s_wait_tensorcnt(0);
  __syncthreads();

  const int base = lane * CHUNK;
  const int nseg = (lane == NCHUNK - 1) ? (CHUNK - 1) : CHUNK;  // last chunk: 63 segments

  // padded LDS indexing: element l lives at (l/64)*65 + (l%64)
#define LDSX(l) xs[(((l) >> 6) * LDS_STRIDE) + ((l) & 63)]

  // ---------- pass 1: per-chunk signature ----------
  Sig s = sig_zero();
  {
    float2 pl = mlp2(LDSX(base), w1r, b1r, w2r, b2r);
    for (int t = 0; t < nseg; ++t) {
      float2 pr = mlp2(LDSX(base + t + 1), w1r, b1r, w2r, b2r);
      chen_seg(s, pr.x - pl.x, pr.y - pl.y);
      pl = pr;
    }
  }

  // ---------- intra-wave exclusive Chen prefix scan (associative) ----------
#pragma unroll
  for (int d = 1; d < 32; d <<= 1) {
    Sig o = shfl_up_sig(s, d);
    if (lane >= d) s = chen_comb(o, s);
  }
  Sig carry = shfl_up_sig(s, 1);
  if (lane == 0) carry = sig_zero();

  // ---------- pass 2: replay chunk from prefix, emit projections into LDS ----------
  {
    float2 pl = mlp2(LDSX(base), w1r, b1r, w2r, b2r);
    for (int t = 0; t < nseg; ++t) {
      float2 pr = mlp2(LDSX(base + t + 1), w1r, b1r, w2r, b2r);
      chen_seg(carry, pr.x - pl.x, pr.y - pl.y);
      pl = pr;
      float acc = wo[0] * carry.a1[0] + wo[1] * carry.a1[1];
#pragma unroll
      for (int i = 0; i < 4; ++i) acc = fmaf(wo[2 + i], carry.a2[i], acc);
#pragma unroll
      for (int i = 0; i < 8; ++i) acc = fmaf(wo[6 + i], carry.a3[i], acc);
      os[lane * LDS_STRIDE + t] = acc;   // padded position of element base+t
    }
  }
  __syncthreads();   // orders the ds_stores above before async LDS reads below

  // ---------- harvest: async LDS -> global stores (no VGPR staging, ASYNCcnt) ----------
  {
    const uint64_t orow = (uint64_t)(uintptr_t)out + (uint64_t)brow * NSEG * 4ull;
    const u32 os_base = (u32)(uintptr_t)(&os[0]);          // workgroup-relative LDS byte addr
    for (int r = 0; r < 64; ++r) {
      int l = r * 32 + lane;             // consecutive lanes -> consecutive dwords, coalesced
      if (l < NSEG) {                    // EXEC-masks the single invalid tail lane
        u32 voff = (u32)l << 2;          // byte offset into output row
        u32 lsrc = os_base + ((((u32)l >> 6) * LDS_STRIDE + ((u32)l & 63u)) << 2);
        asm volatile("global_store_async_from_lds_b32 %0, %1, %2"
                     :: "v"(voff), "v"(lsrc), "s"(orow)
                     : "memory");
      }
    }
    asm volatile("s_wait_asynccnt 0x0" ::: "memory");
  }
#undef LDSX
}

extern "C" void kernel_launch(void* const* d_in, const int* in_sizes, int n_in,
                              void* d_out, int out_size, void* d_ws, size_t ws_size,
                              hipStream_t stream) {
  (void)n_in; (void)out_size; (void)d_ws; (void)ws_size;
  const float* x  = (const float*)d_in[0];
  const float* w1 = (const float*)d_in[1];
  const float* b1 = (const float*)d_in[2];
  const float* w2 = (const float*)d_in[3];
  const float* b2 = (const float*)d_in[4];
  const float* wo = (const float*)d_in[5];
  float* out = (float*)d_out;
  const int B = in_sizes[0] / LPATH;     // 2048 rows; one wave32 per row
  sig_scan_kernel<<<dim3(B), dim3(32), 0, stream>>>(x, w1, b1, w2, b2, wo, out);
}